// ConvMatrix2d_40089224741384
// MI455X (gfx1250) — compile-verified
//
#include <hip/hip_runtime.h>

typedef __attribute__((ext_vector_type(2))) float v2f;
typedef __attribute__((ext_vector_type(4))) float v4f;
typedef __attribute__((ext_vector_type(8))) float v8f;

#define BB    32
#define CC    32
#define WSP   14
#define HHH   17             // H*H + 1
#define OD    32
#define NPOS  (WSP * WSP)    // 196
#define CWW   (OD * NPOS)    // 6272
#define XBLK  (NPOS * HHH)   // 3332 floats of x per (b,c)
#define WBLK  (OD * 16)      // 512 floats of weight per c
#define NTILE (NPOS / 4)     // 49 jb iterations
#define STG   (OD * 4 * HHH) // 2176 dwords staged per iteration
#define STG4  (STG / 4)      // 544 float4

// One workgroup per (b,c); 8 wave32s, wave w owns output channels [4w,4w+4).
// votes[o,p,r](ji) = sum_q weight[c,o,p,q] * x[b,c,i,j,q*4+r]
// via V_WMMA_F32_16X16X4_F32:  A(16x4)=weight rows (o_l*4+p), B(4x16)=poses
// cols (ji_l*4+r).  D tiles + acts are staged in LDS and written out as
// fully-coalesced float4 stores (32 runs of 272 contiguous bytes per jb).
__global__ __launch_bounds__(256, 1)
void capsule_votes_kernel(const float* __restrict__ x,
                          const float* __restrict__ w,
                          float* __restrict__ out)
{
    __shared__ float lds_x[XBLK];
    __shared__ float lds_w[WBLK];
    __shared__ __align__(16) float stage[2][STG];

    const int bc  = blockIdx.x;       // b*32 + c
    const int c   = bc & 31;
    const int tid = threadIdx.x;

    // Stage x[b,c] (13.3 KB) and weight[c] (2 KB) into LDS, coalesced.
    {
        const float* xsrc = x + (size_t)bc * XBLK;
        for (int idx = tid; idx < XBLK; idx += 256) lds_x[idx] = xsrc[idx];
        const float* wsrc = w + (size_t)c * WBLK;
        for (int idx = tid; idx < WBLK; idx += 256) lds_w[idx] = wsrc[idx];
    }
    __syncthreads();

    const int lane = tid & 31;
    const int wave = tid >> 5;         // 0..7 -> o-block
    const int hi   = lane >> 4;        // lanes 16-31 hold K=2,3
    const int n    = lane & 15;
    const int q0   = hi * 2;

    // A operand: row M = n = o_local*4 + p, elements K = q0, q0+1 (wave-constant)
    const int o_l = n >> 2;
    const int p   = n & 3;
    v2f a;
    {
        const int o_row = wave * 4 + o_l;
        a.x = lds_w[o_row * 16 + p * 4 + q0];
        a.y = lds_w[o_row * 16 + p * 4 + q0 + 1];
    }

    // B operand: column N = n = ji_local*4 + r
    const int ji_l = n >> 2;
    const int r    = n & 3;

    // jb-invariant read-out assignments (float4 granularity):
    // float4 idx -> (o = idx/17, c4 = idx%17); global = obase4 + o*833 + jb*17 + c4
    int ro_o[3], ro_c4[3];
    #pragma unroll
    for (int k = 0; k < 3; ++k) {
        const int idx = tid + k * 256;
        ro_o[k]  = idx / HHH;
        ro_c4[k] = idx - ro_o[k] * HHH;
    }

    // act staging assignment (threads 0..127): one (o, ji_l) slot each
    const int act_o    = tid >> 2;
    const int act_ji_l = tid & 3;

    v4f* out4 = (v4f*)out;
    const size_t obase4 = (size_t)bc * (CWW * HHH / 4);  // 26656 float4 per (b,c)

    // Preload B for jb = 0 (software pipeline).
    v2f bv;
    {
        const int ji = ji_l;
        const int j  = ji / WSP;
        const int i  = ji - j * WSP;
        const int xo = (i * WSP + j) * HHH + r;   // poses spatially transposed
        bv.x = lds_x[xo + q0 * 4];
        bv.y = lds_x[xo + (q0 + 1) * 4];
    }

    v8f czero = {};

    for (int jb = 0; jb < NTILE; ++jb) {
        v8f d = __builtin_amdgcn_wmma_f32_16x16x4_f32(
            false, a, false, bv, (short)0, czero, false, false);

        // Preload next B while the WMMA result is staged.
        if (jb + 1 < NTILE) {
            const int ji = (jb + 1) * 4 + ji_l;
            const int j  = ji / WSP;
            const int i  = ji - j * WSP;
            const int xo = (i * WSP + j) * HHH + r;
            bv.x = lds_x[xo + q0 * 4];
            bv.y = lds_x[xo + (q0 + 1) * 4];
        }

        float* st = stage[jb & 1];

        // Stage votes: D vgpr v holds row M = v + 8*hi at column N = lane&15.
        #pragma unroll
        for (int v = 0; v < 8; ++v) {
            const int M  = v + hi * 8;
            const int o  = wave * 4 + (M >> 2);
            const int pp = M & 3;
            st[o * 68 + ji_l * HHH + pp * 4 + r] = d[v];
        }
        // Stage acts (e = 16), broadcast over o.
        if (tid < 128) {
            const int ji = jb * 4 + act_ji_l;
            const int j  = ji / WSP;
            const int i  = ji - j * WSP;
            st[act_o * 68 + act_ji_l * HHH + 16] = lds_x[(i * WSP + j) * HHH + 16];
        }
        __syncthreads();

        // Coalesced write-out: 544 float4 = 32 runs of 272B (16B-aligned).
        const v4f* st4 = (const v4f*)st;
        #pragma unroll
        for (int k = 0; k < 2; ++k) {
            const int idx = tid + k * 256;
            out4[obase4 + (size_t)ro_o[k] * 833 + jb * HHH + ro_c4[k]] = st4[idx];
        }
        if (tid < 32) {
            const int idx = tid + 512;
            out4[obase4 + (size_t)ro_o[2] * 833 + jb * HHH + ro_c4[2]] = st4[idx];
        }
        // Ping-pong buffering: next iteration writes the other buffer, so no
        // second barrier is needed (WAR separated by the following barrier).
    }
}

extern "C" void kernel_launch(void* const* d_in, const int* in_sizes, int n_in,
                              void* d_out, int out_size, void* d_ws, size_t ws_size,
                              hipStream_t stream) {
    const float* x  = (const float*)d_in[0];   // (32,32,14,14,17) f32
    const float* wt = (const float*)d_in[1];   // (32,1,1,32,4,4)  f32
    float* out = (float*)d_out;                // (32,32,6272,17)  f32

    dim3 grid(BB * CC), block(256);
    hipLaunchKernelGGL(capsule_votes_kernel, grid, block, 0, stream, x, wt, out);
}